// GCL_GCN_22771916603815
// MI455X (gfx1250) — compile-verified
//
#include <hip/hip_runtime.h>

#define FEAT 128
#define HID1 128
#define HID2 64

typedef __attribute__((ext_vector_type(2))) float v2f;
typedef __attribute__((ext_vector_type(8))) float v8f;

// ---------------------------------------------------------------------------
// Degree handling
// ---------------------------------------------------------------------------
__global__ void deg_init_kernel(float* __restrict__ deg, int n) {
    int i = blockIdx.x * blockDim.x + threadIdx.x;
    if (i < n) deg[i] = 1.0f;  // self-loop
}

__global__ void deg_accum_kernel(const int* __restrict__ dst,
                                 float* __restrict__ deg, int nE) {
    int e = blockIdx.x * blockDim.x + threadIdx.x;
    if (e < nE) atomicAdd(&deg[dst[e]], 1.0f);
}

__global__ void deg_to_dinv_kernel(float* __restrict__ deg, int n) {
    int i = blockIdx.x * blockDim.x + threadIdx.x;
    if (i < n) deg[i] = rsqrtf(fmaxf(deg[i], 1.0f));
}

// ---------------------------------------------------------------------------
// fp32 WMMA GEMM: D[M x NCOLS] = A[M x 128] * W[128 x NCOLS]
// One wave computes one 16x16 tile using V_WMMA_F32_16X16X4_F32, K loop of 32.
// Block = (NCOLS/16) waves, covering one 16-row strip across all columns.
// RELU_IN applies relu() to A on load (fuses layer-1 activation into GEMM-2).
// ---------------------------------------------------------------------------
template <int NCOLS, bool RELU_IN>
__global__ __launch_bounds__((NCOLS / 16) * 32)
void gemm_wmma_f32(const float* __restrict__ A, const float* __restrict__ W,
                   float* __restrict__ D) {
    const int lane = threadIdx.x & 31;
    const int wave = threadIdx.x >> 5;
    const int m0   = blockIdx.x * 16;
    const int n0   = wave * 16;
    const int mn   = lane & 15;        // A-row within tile / B,D-column within tile
    const int kh   = lane >> 4;        // 0: K pair {0,1};  1: K pair {2,3}
    const int klo  = kh * 2;

    const float* arow = A + (size_t)(m0 + mn) * FEAT;
    v8f c = {};

    #pragma unroll 8
    for (int k = 0; k < FEAT; k += 4) {
        // A 16x4 tile: lane holds A[mn][k+klo], A[mn][k+klo+1]  (8B aligned)
        v2f a = *reinterpret_cast<const v2f*>(arow + k + klo);
        if (RELU_IN) {
            a.x = fmaxf(a.x, 0.0f);
            a.y = fmaxf(a.y, 0.0f);
        }
        // B 4x16 tile: lane holds W[k+klo][n0+mn], W[k+klo+1][n0+mn]
        v2f b;
        b.x = W[(size_t)(k + klo)     * NCOLS + n0 + mn];
        b.y = W[(size_t)(k + klo + 1) * NCOLS + n0 + mn];
        c = __builtin_amdgcn_wmma_f32_16x16x4_f32(
                /*neg_a=*/false, a, /*neg_b=*/false, b,
                /*c_mod=*/(short)0, c, /*reuse_a=*/false, /*reuse_b=*/false);
    }

    // C/D layout: VGPR v -> row (v + 8*kh), col mn
    const int mo = m0 + (kh << 3);
    #pragma unroll
    for (int v = 0; v < 8; ++v) {
        D[(size_t)(mo + v) * NCOLS + n0 + mn] = c[v];
    }
}

// ---------------------------------------------------------------------------
// Per-node init: out[i,f] = dinv[i]^2 * h[i,f] + bias[f]   (self-loop + bias)
// ---------------------------------------------------------------------------
template <int F>
__global__ void node_init_kernel(const float* __restrict__ h,
                                 const float* __restrict__ dinv,
                                 const float* __restrict__ bias,
                                 float* __restrict__ out, int n) {
    int t = blockIdx.x * blockDim.x + threadIdx.x;
    if (t >= n * F) return;
    int i = t / F;
    int f = t & (F - 1);
    float di = dinv[i];
    out[t] = di * di * h[t] + bias[f];
}

// ---------------------------------------------------------------------------
// Edge scatter, F=128: one wave per edge, float4 gather + 4 f32 atomics/lane.
// ---------------------------------------------------------------------------
__global__ void edge_agg128_kernel(const int* __restrict__ src,
                                   const int* __restrict__ dst,
                                   const float* __restrict__ dinv,
                                   const float* __restrict__ h,
                                   float* __restrict__ out, int nE) {
    const int lane = threadIdx.x & 31;
    const int e = blockIdx.x * (blockDim.x >> 5) + (threadIdx.x >> 5);
    if (e >= nE) return;
    const int s = src[e];
    const int d = dst[e];
    const float norm = dinv[s] * dinv[d];
    const float4 hv = *reinterpret_cast<const float4*>(h + (size_t)s * 128 + lane * 4);
    float* od = out + (size_t)d * 128 + lane * 4;
    atomicAdd(od + 0, norm * hv.x);
    atomicAdd(od + 1, norm * hv.y);
    atomicAdd(od + 2, norm * hv.z);
    atomicAdd(od + 3, norm * hv.w);
}

// Edge scatter, F=64: one wave per edge, float2 gather + 2 f32 atomics/lane.
__global__ void edge_agg64_kernel(const int* __restrict__ src,
                                  const int* __restrict__ dst,
                                  const float* __restrict__ dinv,
                                  const float* __restrict__ h,
                                  float* __restrict__ out, int nE) {
    const int lane = threadIdx.x & 31;
    const int e = blockIdx.x * (blockDim.x >> 5) + (threadIdx.x >> 5);
    if (e >= nE) return;
    const int s = src[e];
    const int d = dst[e];
    const float norm = dinv[s] * dinv[d];
    const float2 hv = *reinterpret_cast<const float2*>(h + (size_t)s * 64 + lane * 2);
    float* od = out + (size_t)d * 64 + lane * 2;
    atomicAdd(od + 0, norm * hv.x);
    atomicAdd(od + 1, norm * hv.y);
}

// ---------------------------------------------------------------------------
// Launcher
// ---------------------------------------------------------------------------
extern "C" void kernel_launch(void* const* d_in, const int* in_sizes, int n_in,
                              void* d_out, int out_size, void* d_ws, size_t ws_size,
                              hipStream_t stream) {
    const float* x  = (const float*)d_in[0];
    const int*   ei = (const int*)  d_in[1];
    const float* W1 = (const float*)d_in[2];
    const float* b1 = (const float*)d_in[3];
    const float* W2 = (const float*)d_in[4];
    const float* b2 = (const float*)d_in[5];
    float* out = (float*)d_out;

    const int N = in_sizes[0] / FEAT;   // 50000 (divisible by 16)
    const int E = in_sizes[1] / 2;      // 800000
    const int* src = ei;
    const int* dst = ei + E;

    // Workspace partition (floats)
    float* ws   = (float*)d_ws;
    size_t off  = ((size_t)N + 255) & ~(size_t)255;
    float* dinv = ws;                       // [N]   deg -> rsqrt(deg) in place
    float* h1   = ws + off;                 // [N,128] X@W1
    float* agg1 = h1 + (size_t)N * HID1;    // [N,128] aggregated layer 1 (pre-relu)
    float* h2   = agg1 + (size_t)N * HID1;  // [N,64]  relu(agg1)@W2

    const int T = 256;

    // 1) degrees (self-loop included) -> dinv
    deg_init_kernel<<<(N + T - 1) / T, T, 0, stream>>>(dinv, N);
    deg_accum_kernel<<<(E + T - 1) / T, T, 0, stream>>>(dst, dinv, E);
    deg_to_dinv_kernel<<<(N + T - 1) / T, T, 0, stream>>>(dinv, N);

    // 2) h1 = X @ W1  (fp32 WMMA)
    gemm_wmma_f32<HID1, false><<<N / 16, (HID1 / 16) * 32, 0, stream>>>(x, W1, h1);

    // 3) agg1 = dinv^2*h1 + b1, then scatter edges
    node_init_kernel<HID1><<<((size_t)N * HID1 + T - 1) / T, T, 0, stream>>>(
        h1, dinv, b1, agg1, N);
    edge_agg128_kernel<<<(E + 7) / 8, T, 0, stream>>>(src, dst, dinv, h1, agg1, E);

    // 4) h2 = relu(agg1) @ W2  (fp32 WMMA, relu fused into A load)
    gemm_wmma_f32<HID2, true><<<N / 16, (HID2 / 16) * 32, 0, stream>>>(agg1, W2, h2);

    // 5) out = dinv^2*h2 + b2, then scatter edges
    node_init_kernel<HID2><<<((size_t)N * HID2 + T - 1) / T, T, 0, stream>>>(
        h2, dinv, b2, out, N);
    edge_agg64_kernel<<<(E + 7) / 8, T, 0, stream>>>(src, dst, dinv, h2, out, E);
}